// ResGNN_53214644798105
// MI455X (gfx1250) — compile-verified
//
#include <hip/hip_runtime.h>
#include <hip/hip_bf16.h>
#include <cstdint>
#include <cstddef>

// ---------------------------------------------------------------------------
// ResGNN forward for MI455X (gfx1250, wave32).
// GEMMs use V_WMMA_F32_16X16X4_F32 (exact f32, matches reference numerics).
// Edge aggregation uses hardware f32 global atomics (global_atomic_add_f32).
// ---------------------------------------------------------------------------

#define N_   100000
#define E_   800000
#define H_   128
#define B_   512
#define C_   10
#define L_   3
#define EPS_ 1e-5f

typedef __attribute__((ext_vector_type(2))) float v2f;
typedef __attribute__((ext_vector_type(8))) float v8f;

// ------------------------------- small kernels ------------------------------

__global__ void deg_kernel(const int* __restrict__ dst, float* __restrict__ deg, int e) {
    int i = blockIdx.x * blockDim.x + threadIdx.x;
    if (i < e) unsafeAtomicAdd(&deg[dst[i]], 1.0f);
}

__global__ void dinv_kernel(float* __restrict__ deg, int n) {
    int i = blockIdx.x * blockDim.x + threadIdx.x;
    if (i < n) deg[i] = __frsqrt_rn(deg[i] + 1.0f);   // dinv in place
}

// per-feature sums: block = 128 threads (one feature each), 256 rows per block
__global__ void bn_stats_kernel(const float* __restrict__ x, int nrows,
                                float* __restrict__ stats) {
    int f  = threadIdx.x;                 // 0..127
    int r0 = blockIdx.x * 256;
    int r1 = min(nrows, r0 + 256);
    float s = 0.f, s2 = 0.f;
    for (int r = r0; r < r1; ++r) {
        float v = x[(size_t)r * H_ + f];
        s += v; s2 += v * v;
    }
    unsafeAtomicAdd(&stats[f],        s);
    unsafeAtomicAdd(&stats[H_ + f],   s2);
}

// stats[0..127]=sum, [128..255]=sumsq  ->  [256..383]=scale, [384..511]=shift
__global__ void bn_finalize_kernel(float* __restrict__ stats,
                                   const float* __restrict__ gam,
                                   const float* __restrict__ bet, int nrows) {
    int f = threadIdx.x;
    float inv_n = 1.0f / (float)nrows;
    float mu  = stats[f] * inv_n;
    float var = stats[H_ + f] * inv_n - mu * mu;       // biased, as torch BN
    float sc  = gam[f] * rsqrtf(var + EPS_);
    stats[256 + f] = sc;
    stats[384 + f] = bet[f] - mu * sc;
}

__global__ void bn_apply_kernel(const float* __restrict__ x,
                                const float* __restrict__ scale,
                                const float* __restrict__ shift,
                                float* __restrict__ y, int nrows) {
    int i = blockIdx.x * blockDim.x + threadIdx.x;     // float4 index
    int total = nrows * (H_ / 4);
    if (i >= total) return;
    int c4 = i & (H_ / 4 - 1);
    float4 v  = ((const float4*)x)[i];
    float4 sc = ((const float4*)scale)[c4];
    float4 sh = ((const float4*)shift)[c4];
    v.x = v.x * sc.x + sh.x;  v.y = v.y * sc.y + sh.y;
    v.z = v.z * sc.z + sh.z;  v.w = v.w * sc.w + sh.w;
    ((float4*)y)[i] = v;
}

// agg = xw * dinv^2   (self-loop term; also serves as the scatter init)
__global__ void selfloop_init_kernel(const float* __restrict__ xw,
                                     const float* __restrict__ dinv,
                                     float* __restrict__ agg, int nrows) {
    int i = blockIdx.x * blockDim.x + threadIdx.x;
    int total = nrows * (H_ / 4);
    if (i >= total) return;
    int row = i >> 5;                                  // /(H_/4)
    float d = dinv[row];
    float c = d * d;
    float4 v = ((const float4*)xw)[i];
    v.x *= c; v.y *= c; v.z *= c; v.w *= c;
    ((float4*)agg)[i] = v;
}

// one wave per edge, one float4 of features per lane -> global f32 atomics
__global__ void edge_scatter_kernel(const int* __restrict__ src,
                                    const int* __restrict__ dst,
                                    const float* __restrict__ dinv,
                                    const float* __restrict__ xw,
                                    float* __restrict__ agg, int e) {
    int gid = blockIdx.x * blockDim.x + threadIdx.x;
    int eid = gid >> 5;
    if (eid >= e) return;
    int l = gid & 31;
    int s = src[eid], d = dst[eid];
    float coef = dinv[s] * dinv[d];
    float4 v = *(const float4*)(xw + (size_t)s * H_ + l * 4);
    float* a = agg + (size_t)d * H_ + l * 4;
    unsafeAtomicAdd(a + 0, v.x * coef);
    unsafeAtomicAdd(a + 1, v.y * coef);
    unsafeAtomicAdd(a + 2, v.z * coef);
    unsafeAtomicAdd(a + 3, v.w * coef);
}

// h = relu(agg + bias)
__global__ void conv_finish_kernel(const float* __restrict__ agg,
                                   const float* __restrict__ bias,
                                   float* __restrict__ h, int nrows) {
    int i = blockIdx.x * blockDim.x + threadIdx.x;
    int total = nrows * (H_ / 4);
    if (i >= total) return;
    int c4 = i & (H_ / 4 - 1);
    float4 b = ((const float4*)bias)[c4];
    float4 v = ((const float4*)agg)[i];
    v.x = fmaxf(v.x + b.x, 0.f);  v.y = fmaxf(v.y + b.y, 0.f);
    v.z = fmaxf(v.z + b.z, 0.f);  v.w = fmaxf(v.w + b.w, 0.f);
    ((float4*)h)[i] = v;
}

// global_add_pool: g[batch[i]] += h[i]
__global__ void pool_kernel(const float* __restrict__ h,
                            const int* __restrict__ batch,
                            float* __restrict__ g, int nrows) {
    int gid = blockIdx.x * blockDim.x + threadIdx.x;
    int row = gid >> 5;
    if (row >= nrows) return;
    int l = gid & 31;
    int b = batch[row];
    float4 v = *(const float4*)(h + (size_t)row * H_ + l * 4);
    float* gp = g + (size_t)b * H_ + l * 4;
    unsafeAtomicAdd(gp + 0, v.x);
    unsafeAtomicAdd(gp + 1, v.y);
    unsafeAtomicAdd(gp + 2, v.z);
    unsafeAtomicAdd(gp + 3, v.w);
}

// logits = g @ W_cls + b_cls; log_softmax. One thread per graph (512 total).
__global__ void cls_kernel(const float* __restrict__ g,
                           const float* __restrict__ W,
                           const float* __restrict__ b,
                           float* __restrict__ out, int nrows) {
    int row = blockIdx.x * blockDim.x + threadIdx.x;
    if (row >= nrows) return;
    const float* gr = g + (size_t)row * H_;
    float l[C_];
    #pragma unroll
    for (int c = 0; c < C_; ++c) l[c] = b[c];
    for (int k = 0; k < H_; ++k) {
        float gv = gr[k];
        #pragma unroll
        for (int c = 0; c < C_; ++c) l[c] += gv * W[k * C_ + c];
    }
    float m = l[0];
    #pragma unroll
    for (int c = 1; c < C_; ++c) m = fmaxf(m, l[c]);
    float s = 0.f;
    #pragma unroll
    for (int c = 0; c < C_; ++c) s += expf(l[c] - m);
    float lse = m + logf(s);
    #pragma unroll
    for (int c = 0; c < C_; ++c) out[(size_t)row * C_ + c] = l[c] - lse;
}

// ------------------------------ WMMA f32 GEMM -------------------------------
// C[nrows,128] = A[nrows,128] @ W[128,128]  (+bias) (+relu)
// 256 threads = 8 waves; each wave: 16-row strip x all 8 column tiles.
// f32 A-frag (16x4): lanes 0-15 hold K={k0,k0+1}, lanes 16-31 hold K={k0+2,k0+3}.
// f32 B-frag (4x16): same K split across lane halves, N = lane&15.
// f32 C/D (16x16, v8f): lanes 0-15 -> M=r, lanes 16-31 -> M=8+r.
__global__ __launch_bounds__(256)
void gemm128_wmma_kernel(const float* __restrict__ A,
                         const float* __restrict__ W,
                         const float* __restrict__ bias,
                         float* __restrict__ C, int nrows, int relu) {
    __shared__ float sW[H_ * H_];                       // 64 KB
    // stage W through LDS (float4)
    {
        const float4* W4 = (const float4*)W;
        float4* s4 = (float4*)sW;
        #pragma unroll
        for (int i = threadIdx.x; i < H_ * H_ / 4; i += 256) s4[i] = W4[i];
    }
    __syncthreads();

    const int lane = threadIdx.x & 31;
    const int wave = threadIdx.x >> 5;
    const int half = lane >> 4;                         // 0: K lo pair, 1: K hi pair
    const int l16  = lane & 15;
    const int koff = half * 2;
    const int rowBase = blockIdx.x * 128 + wave * 16;

    int rowA = rowBase + l16;
    if (rowA >= nrows) rowA = nrows - 1;                // clamp reads; stores guarded
    const float* Arow = A + (size_t)rowA * H_;

    v8f acc[8];
    #pragma unroll
    for (int t = 0; t < 8; ++t) acc[t] = (v8f)0.0f;

    #pragma unroll
    for (int k0 = 0; k0 < H_; k0 += 4) {
        float2 a2 = *(const float2*)(Arow + k0 + koff);
        v2f a; a.x = a2.x; a.y = a2.y;
        const float* wk = sW + (k0 + koff) * H_ + l16;
        #pragma unroll
        for (int t = 0; t < 8; ++t) {
            v2f b; b.x = wk[t * 16]; b.y = wk[H_ + t * 16];
            acc[t] = __builtin_amdgcn_wmma_f32_16x16x4_f32(
                false, a, false, b, (short)0, acc[t], false, false);
        }
    }

    #pragma unroll
    for (int t = 0; t < 8; ++t) {
        int col = t * 16 + l16;
        float bv = bias ? bias[col] : 0.0f;
        #pragma unroll
        for (int r = 0; r < 8; ++r) {
            int row = rowBase + half * 8 + r;
            if (row < nrows) {
                float v = acc[t][r] + bv;
                if (relu) v = fmaxf(v, 0.0f);
                C[(size_t)row * H_ + col] = v;
            }
        }
    }
}

// ------------------------------- launch glue --------------------------------

static inline void run_bn(const float* x, const float* gam, const float* bet,
                          float* y, float* stats, int nrows, hipStream_t st) {
    hipMemsetAsync(stats, 0, 256 * sizeof(float), st);
    int sgrid = (nrows + 255) / 256;
    bn_stats_kernel<<<sgrid, 128, 0, st>>>(x, nrows, stats);
    bn_finalize_kernel<<<1, 128, 0, st>>>(stats, gam, bet, nrows);
    int agrid = (nrows * (H_ / 4) + 255) / 256;
    bn_apply_kernel<<<agrid, 256, 0, st>>>(x, stats + 256, stats + 384, y, nrows);
}

extern "C" void kernel_launch(void* const* d_in, const int* in_sizes, int n_in,
                              void* d_out, int out_size, void* d_ws, size_t ws_size,
                              hipStream_t stream) {
    const float* x         = (const float*)d_in[0];
    const int*   ei        = (const int*)  d_in[1];
    const int*   batch     = (const int*)  d_in[2];
    const float* bnf_g     = (const float*)d_in[3];
    const float* bnf_b     = (const float*)d_in[4];
    const float* W_feat    = (const float*)d_in[5];
    const float* cbn_g     = (const float*)d_in[6];
    const float* cbn_b     = (const float*)d_in[7];
    const float* conv_W    = (const float*)d_in[8];
    const float* conv_bias = (const float*)d_in[9];
    const float* bnfc_g    = (const float*)d_in[10];
    const float* bnfc_b    = (const float*)d_in[11];
    const float* W_fc      = (const float*)d_in[12];
    const float* b_fc      = (const float*)d_in[13];
    const float* bnh_g     = (const float*)d_in[14];
    const float* bnh_b     = (const float*)d_in[15];
    const float* W_cls     = (const float*)d_in[16];
    const float* b_cls     = (const float*)d_in[17];
    float* out = (float*)d_out;

    const int* src = ei;        // edge_index[0, :]
    const int* dst = ei + E_;   // edge_index[1, :]

    // workspace carve-up (~155 MB): 3 node buffers + dinv + stats + 2 pooled
    float* ws    = (float*)d_ws;
    float* h     = ws;                               // N x 128
    float* t1    = h    + (size_t)N_ * H_;           // N x 128 (bn out)
    float* t2    = t1   + (size_t)N_ * H_;           // N x 128 (xw)
    float* dinv  = t2   + (size_t)N_ * H_;           // N
    float* stats = dinv + N_;                        // 512
    float* gbuf  = stats + 512;                      // B x 128
    float* gtmp  = gbuf + (size_t)B_ * H_;           // B x 128

    const int elemGridN = (N_ * (H_ / 4) + 255) / 256;
    const int gemmGridN = (N_ + 127) / 128;

    // degrees + dinv (shared by all conv layers)
    hipMemsetAsync(dinv, 0, N_ * sizeof(float), stream);
    deg_kernel<<<(E_ + 255) / 256, 256, 0, stream>>>(dst, dinv, E_);
    dinv_kernel<<<(N_ + 255) / 256, 256, 0, stream>>>(dinv, N_);

    // h = relu(bn_feat(x) @ W_feat)
    run_bn(x, bnf_g, bnf_b, t1, stats, N_, stream);
    gemm128_wmma_kernel<<<gemmGridN, 256, 0, stream>>>(t1, W_feat, nullptr, h, N_, 1);

    // 3 GCN conv layers
    for (int i = 0; i < L_; ++i) {
        run_bn(h, cbn_g + i * H_, cbn_b + i * H_, t1, stats, N_, stream);
        gemm128_wmma_kernel<<<gemmGridN, 256, 0, stream>>>(
            t1, conv_W + (size_t)i * H_ * H_, nullptr, t2, N_, 0);   // xw -> t2
        selfloop_init_kernel<<<elemGridN, 256, 0, stream>>>(t2, dinv, t1, N_); // agg init -> t1
        edge_scatter_kernel<<<(E_ * 32 + 255) / 256, 256, 0, stream>>>(
            src, dst, dinv, t2, t1, E_);
        conv_finish_kernel<<<elemGridN, 256, 0, stream>>>(
            t1, conv_bias + i * H_, h, N_);                          // h = relu(agg+bias)
    }

    // global_add_pool -> gbuf [B,128]
    hipMemsetAsync(gbuf, 0, (size_t)B_ * H_ * sizeof(float), stream);
    pool_kernel<<<(N_ * 32 + 255) / 256, 256, 0, stream>>>(h, batch, gbuf, N_);

    // g = relu(bn_fc(g) @ W_fc + b_fc)
    run_bn(gbuf, bnfc_g, bnfc_b, gtmp, stats, B_, stream);
    gemm128_wmma_kernel<<<(B_ + 127) / 128, 256, 0, stream>>>(
        gtmp, W_fc, b_fc, gbuf, B_, 1);

    // g = bn_hid(g); logits + log_softmax
    run_bn(gbuf, bnh_g, bnh_b, gtmp, stats, B_, stream);
    cls_kernel<<<(B_ + 127) / 128, 128, 0, stream>>>(gtmp, W_cls, b_cls, out, B_);
}